// GATv2_29712583754281
// MI455X (gfx1250) — compile-verified
//
#include <hip/hip_runtime.h>
#include <hip/hip_bf16.h>
#include <stdint.h>

// ---------------------------------------------------------------------------
// GATv2 3-layer pipeline for MI455X (gfx1250), fp32 end to end.
// GEMMs use V_WMMA_F32_16X16X4_F32 (one wave -> one 16x16 C tile).
// Edge phase: 3 memory-bound passes with f32/u32 atomics.
// ---------------------------------------------------------------------------

typedef __attribute__((ext_vector_type(2))) float v2f;
typedef __attribute__((ext_vector_type(8))) float v8f;

#define HC 256
#define NHEAD 4
#define CH 64

// ---------------------------------------------------------------------------
// Generic fill
// ---------------------------------------------------------------------------
__global__ void fill_f32(float* __restrict__ p, int n, float v) {
    int i = blockIdx.x * blockDim.x + threadIdx.x;
    if (i < n) p[i] = v;
}

// ---------------------------------------------------------------------------
// C[M,Nc] = A[M,K] @ W[K,Nc] + bias   (optional ReLU)
// grid = (Nc/16, M/16), block = 32 (one wave per 16x16 tile).
// A row-major [M,K], W row-major [K,Nc].
//
// VGPR layouts per CDNA5 ISA 7.12.2 (wave32):
//   A 16x4 : lane<16: M=lane, K={0,1} ; lane>=16: M=lane-16, K={2,3}
//   B 4x16 : lane<16: N=lane, K={0,1} ; lane>=16: N=lane-16, K={2,3}
//   C 16x16: vgpr r, lane<16: M=r, N=lane ; lane>=16: M=r+8, N=lane-16
// ---------------------------------------------------------------------------
__global__ __launch_bounds__(32) void wmma_gemm_bias(
    const float* __restrict__ A, const float* __restrict__ W,
    const float* __restrict__ bias, float* __restrict__ C,
    int K, int Nc, int act)
{
    const int lane = threadIdx.x;
    const int half = lane >> 4;      // 0 or 1
    const int l16  = lane & 15;
    const int n0   = blockIdx.x * 16;
    const int m0   = blockIdx.y * 16;

    v8f acc = {};
    const float* Arow = A + (size_t)(m0 + l16) * K;

    for (int k = 0; k < K; k += 4) {
        const int kk = k + 2 * half;
        v2f a;
        a[0] = Arow[kk];
        a[1] = Arow[kk + 1];
        v2f b;
        b[0] = W[(size_t)kk * Nc + n0 + l16];
        b[1] = W[(size_t)(kk + 1) * Nc + n0 + l16];
        acc = __builtin_amdgcn_wmma_f32_16x16x4_f32(
            /*neg_a=*/false, a, /*neg_b=*/false, b,
            /*c_mod=*/(short)0, acc, /*reuse_a=*/false, /*reuse_b=*/false);
    }

    const int col = n0 + l16;
    const float bv = bias[col];
#pragma unroll
    for (int r = 0; r < 8; ++r) {
        const int row = m0 + 8 * half + r;
        float v = acc[r] + bv;
        if (act) v = fmaxf(v, 0.0f);
        C[(size_t)row * Nc + col] = v;
    }
}

// ---------------------------------------------------------------------------
// Monotonic float<->uint encoding for atomicMax-based float max.
// ---------------------------------------------------------------------------
__device__ __forceinline__ unsigned enc_f32(float x) {
    unsigned u = __float_as_uint(x);
    return (u & 0x80000000u) ? ~u : (u | 0x80000000u);
}
__device__ __forceinline__ float dec_f32(unsigned u) {
    unsigned b = (u & 0x80000000u) ? (u ^ 0x80000000u) : ~u;
    return __uint_as_float(b);
}

// ---------------------------------------------------------------------------
// Pass 1: alpha[e][h] = att[h] . leaky_relu(xl[src] + xr[dst]); m = seg-max.
// One wave per edge.
// ---------------------------------------------------------------------------
__global__ void edge_alpha_kernel(
    const float* __restrict__ xl, const float* __restrict__ xr,
    const float* __restrict__ att,
    const int* __restrict__ src, const int* __restrict__ dst,
    float* __restrict__ alpha, unsigned* __restrict__ menc, int E)
{
    const int gid  = blockIdx.x * blockDim.x + threadIdx.x;
    const int e    = gid >> 5;
    const int lane = gid & 31;
    if (e >= E) return;

    const int s = src[e], d = dst[e];
    const float* pl = xl + (size_t)s * HC;
    const float* pr = xr + (size_t)d * HC;

#pragma unroll
    for (int h = 0; h < NHEAD; ++h) {
        float acc = 0.0f;
#pragma unroll
        for (int j = 0; j < 2; ++j) {
            const int c = h * CH + lane + j * 32;
            float v = pl[c] + pr[c];
            v = (v > 0.0f) ? v : 0.2f * v;   // leaky_relu(0.2)
            acc += v * att[c];
        }
        for (int off = 16; off; off >>= 1) acc += __shfl_down(acc, off);
        if (lane == 0) {
            alpha[(size_t)e * NHEAD + h] = acc;
            atomicMax(&menc[d * NHEAD + h], enc_f32(acc));
        }
    }
}

// ---------------------------------------------------------------------------
// Pass 2: a = exp(alpha - m[dst]); s[dst] += a.  One thread per (e,h).
// ---------------------------------------------------------------------------
__global__ void edge_exp_kernel(
    const int* __restrict__ dst, const unsigned* __restrict__ menc,
    float* __restrict__ alpha, float* __restrict__ ssum, int EH)
{
    const int i = blockIdx.x * blockDim.x + threadIdx.x;
    if (i >= EH) return;
    const int e = i >> 2, h = i & 3;
    const int d = dst[e];
    const float m = dec_f32(menc[d * NHEAD + h]);
    const float a = expf(alpha[i] - m);
    alpha[i] = a;
    atomicAdd(&ssum[d * NHEAD + h], a);
}

// ---------------------------------------------------------------------------
// Pass 3: accum[dst] += (a/s[dst]) * xl[src].  One wave per edge.
// ---------------------------------------------------------------------------
__global__ void edge_scatter_kernel(
    const float* __restrict__ xl,
    const int* __restrict__ src, const int* __restrict__ dst,
    const float* __restrict__ alpha, const float* __restrict__ ssum,
    float* __restrict__ accum, int E)
{
    const int gid  = blockIdx.x * blockDim.x + threadIdx.x;
    const int e    = gid >> 5;
    const int lane = gid & 31;
    if (e >= E) return;

    const int s = src[e], d = dst[e];
    const float* pl = xl + (size_t)s * HC;
    float* pa = accum + (size_t)d * HC;

    float w[NHEAD];
#pragma unroll
    for (int h = 0; h < NHEAD; ++h)
        w[h] = alpha[(size_t)e * NHEAD + h] / ssum[d * NHEAD + h];

#pragma unroll
    for (int j = 0; j < 8; ++j) {
        const int c = lane + j * 32;
        atomicAdd(&pa[c], w[c >> 6] * pl[c]);
    }
}

// ---------------------------------------------------------------------------
// Finalize (in place on accum): h = elu(layer_norm(accum + cb + res)).
// One 256-thread block per node.
// ---------------------------------------------------------------------------
__global__ __launch_bounds__(HC) void node_finalize_kernel(
    float* __restrict__ acc, const float* __restrict__ res,
    const float* __restrict__ cb,
    const float* __restrict__ g, const float* __restrict__ b)
{
    const int n = blockIdx.x;
    const int c = threadIdx.x;
    const int lane = c & 31, wid = c >> 5;

    __shared__ float red[8];
    __shared__ float mu_s, rstd_s;

    float val = acc[(size_t)n * HC + c] + cb[c] + res[(size_t)n * HC + c];

    float s = val;
    for (int off = 16; off; off >>= 1) s += __shfl_down(s, off);
    if (lane == 0) red[wid] = s;
    __syncthreads();
    if (c == 0) {
        float t = 0.0f;
#pragma unroll
        for (int i = 0; i < 8; ++i) t += red[i];
        mu_s = t * (1.0f / HC);
    }
    __syncthreads();
    const float mu = mu_s;

    float dv = val - mu;
    float q = dv * dv;
    for (int off = 16; off; off >>= 1) q += __shfl_down(q, off);
    if (lane == 0) red[wid] = q;
    __syncthreads();
    if (c == 0) {
        float t = 0.0f;
#pragma unroll
        for (int i = 0; i < 8; ++i) t += red[i];
        rstd_s = rsqrtf(t * (1.0f / HC) + 1e-5f);
    }
    __syncthreads();

    float y = dv * rstd_s * g[c] + b[c];
    y = (y > 0.0f) ? y : expm1f(y);          // ELU(alpha=1)
    acc[(size_t)n * HC + c] = y;
}

// ---------------------------------------------------------------------------
// Host-side orchestration
// ---------------------------------------------------------------------------
static inline void launch_fill(float* p, int n, float v, hipStream_t st) {
    fill_f32<<<(n + 255) / 256, 256, 0, st>>>(p, n, v);
}

extern "C" void kernel_launch(void* const* d_in, const int* in_sizes, int n_in,
                              void* d_out, int out_size, void* d_ws, size_t ws_size,
                              hipStream_t stream)
{
    const int N = in_sizes[0] / 512;     // 10000
    const int E = in_sizes[1] / 2;       // 330000
    const int IN_DIM = 512;
    const int HID = 64;

    const float* x   = (const float*)d_in[0];
    const int*   ei  = (const int*)d_in[1];
    const int*   src = ei;
    const int*   dst = ei + E;

    const float* ln_g    = (const float*)d_in[20];
    const float* ln_b    = (const float*)d_in[21];
    const float* proj0_W = (const float*)d_in[22];
    const float* proj0_b = (const float*)d_in[23];
    const float* out_W   = (const float*)d_in[24];
    const float* out_b   = (const float*)d_in[25];
    const float* vis_W   = (const float*)d_in[26];
    const float* vis_b   = (const float*)d_in[27];
    const float* txt_W   = (const float*)d_in[28];
    const float* txt_b   = (const float*)d_in[29];

    // Workspace layout (floats)
    const size_t NHC = (size_t)N * HC;
    float* nb0   = (float*)d_ws;
    float* nb1   = nb0 + NHC;
    float* nb2   = nb1 + NHC;
    float* nb3   = nb2 + NHC;
    float* alpha = nb3 + NHC;                     // E * NHEAD
    float* ssum  = alpha + (size_t)E * NHEAD;     // N * NHEAD
    unsigned* menc = (unsigned*)(ssum + (size_t)N * NHEAD); // N * NHEAD

    const size_t need_bytes =
        (4 * NHC + (size_t)E * NHEAD + 2 * (size_t)N * NHEAD) * sizeof(float);
    if (ws_size < need_bytes) return;

    const dim3 wmma_blk(32);
    const int mtiles = N / 16;                    // 625 (N divisible by 16)
    const int edge_blocks = (E * 32 + 255) / 256;
    const int eh_blocks   = (E * NHEAD + 255) / 256;

    // Buffer rotation: xl=nb1, xr=nb2 each layer.
    //   L0: res=nb0 (proj0(x)),  accum/h_out=nb3
    //   L1: h_in/res=nb3,        accum/h_out=nb0
    //   L2: h_in/res=nb0,        accum/h_out=nb3
    const float* hin[3]  = { x,   nb3, nb0 };
    const int    Kin[3]  = { IN_DIM, HC, HC };
    float*       hout[3] = { nb3, nb0, nb3 };

    for (int l = 0; l < 3; ++l) {
        const float* Wl  = (const float*)d_in[2 + l * 6 + 0];
        const float* Wr  = (const float*)d_in[2 + l * 6 + 1];
        const float* bl  = (const float*)d_in[2 + l * 6 + 2];
        const float* br  = (const float*)d_in[2 + l * 6 + 3];
        const float* att = (const float*)d_in[2 + l * 6 + 4];
        const float* cb  = (const float*)d_in[2 + l * 6 + 5];

        const int K = Kin[l];
        const float* resp;
        if (l == 0) {
            // res = x @ proj0_W + proj0_b -> nb0
            wmma_gemm_bias<<<dim3(HC / 16, mtiles), wmma_blk, 0, stream>>>(
                x, proj0_W, proj0_b, nb0, IN_DIM, HC, 0);
            resp = nb0;
        } else {
            resp = hin[l];
        }

        // Node transforms
        wmma_gemm_bias<<<dim3(HC / 16, mtiles), wmma_blk, 0, stream>>>(
            hin[l], Wl, bl, nb1, K, HC, 0);
        wmma_gemm_bias<<<dim3(HC / 16, mtiles), wmma_blk, 0, stream>>>(
            hin[l], Wr, br, nb2, K, HC, 0);

        // Zero segment accumulators
        launch_fill(hout[l], (int)NHC, 0.0f, stream);
        launch_fill(ssum, N * NHEAD, 0.0f, stream);
        launch_fill((float*)menc, N * NHEAD, 0.0f, stream);  // 0u == -inf sentinel

        // Edge phase
        edge_alpha_kernel<<<edge_blocks, 256, 0, stream>>>(
            nb1, nb2, att, src, dst, alpha, menc, E);
        edge_exp_kernel<<<eh_blocks, 256, 0, stream>>>(
            dst, menc, alpha, ssum, E * NHEAD);
        edge_scatter_kernel<<<edge_blocks, 256, 0, stream>>>(
            nb1, src, dst, alpha, ssum, hout[l], E);

        // Residual + LayerNorm + ELU (in place -> new h)
        node_finalize_kernel<<<N, HC, 0, stream>>>(
            hout[l], resp, cb, ln_g + l * HC, ln_b + l * HC);
    }

    // Final projections
    float* outp = (float*)d_out;                   // [N, 64]
    float* visp = outp + (size_t)N * HID;
    float* txtp = visp + (size_t)N * HID;

    wmma_gemm_bias<<<dim3(HID / 16, mtiles), wmma_blk, 0, stream>>>(
        nb3, out_W, out_b, outp, HC, HID, 0);
    wmma_gemm_bias<<<dim3(HID / 16, mtiles), wmma_blk, 0, stream>>>(
        outp, vis_W, vis_b, visp, HID, HID, 1);
    wmma_gemm_bias<<<dim3(HID / 16, mtiles), wmma_blk, 0, stream>>>(
        outp, txt_W, txt_b, txtp, HID, HID, 1);
}